// GAT_31490700214331
// MI455X (gfx1250) — compile-verified
//
#include <hip/hip_runtime.h>
#include <hip/hip_bf16.h>
#include <math.h>

// ---------------------------------------------------------------------------
// GAT forward on gfx1250 (MI455X): WMMA f32 GEMMs (LDS-staged A) +
// atomic edge-softmax scatter.
// ---------------------------------------------------------------------------

typedef __attribute__((ext_vector_type(2))) float v2f;
typedef __attribute__((ext_vector_type(8))) float v8f;

#define N_NODES 50000
#define N_EDGES 600000
#define N_FEAT  256
#define N_HID   128
#define N_CLASS 64
#define HEADS   8
#define LEAKY   0.2f

#define MAX_K   256   // max GEMM K (N_FEAT); LDS A-tile sized for this

// ---------------------------------------------------------------------------
// WMMA tiled GEMM: C[M,Ncols] = A[M,K] @ B[K,Ncols], fp32.
// M % 16 == 0, K % 4 == 0, Ncols == gridDim.y * 128.
//
// Block = 256 threads = 8 waves. All 8 waves share one 16-row m-tile; the
// 16xK A tile (<= 16 KB of the WGP's 320 KB LDS) is staged cooperatively
// once, then every wave reads its A fragments via ds_load_b64 (8-32x less
// global A traffic). Each wave owns one 16-col n-tile; B fragments come from
// global (W0/W1 are 128-256 KB -> L2 resident).
//
// Fragment layouts per CDNA5 ISA 7.12.2 (V_WMMA_F32_16X16X4_F32):
//   A (16x4):  lane<16: M=lane,     K=k0,k0+1 in v0,v1; lane>=16: K=k0+2,k0+3
//   B (4x16):  lane<16: N=lane,     K=k0,k0+1;          lane>=16: K=k0+2,k0+3
//   C/D:       VGPR r -> row M = r + (lane>=16 ? 8 : 0), col N = lane&15
// ---------------------------------------------------------------------------
__global__ __launch_bounds__(256)
void gemm_wmma_f32(const float* __restrict__ A, const float* __restrict__ B,
                   float* __restrict__ C, int K, int Ncols) {
    __shared__ float ldsA[16 * MAX_K];

    const int mtile = blockIdx.x;
    const int wave  = threadIdx.x >> 5;
    const int lane  = threadIdx.x & 31;
    const int ntile = blockIdx.y * 8 + wave;    // wave-uniform, always in range

    // ---- cooperative stage of the 16xK A tile (contiguous rows) ----
    const float* __restrict__ atile = A + (size_t)mtile * 16 * K;
    for (int idx = threadIdx.x; idx < 16 * K; idx += 256)
        ldsA[idx] = atile[idx];
    __syncthreads();

    const int lo16  = lane & 15;
    const int khalf = (lane >> 4) * 2;          // 0 or 2
    const int n     = ntile * 16 + lo16;        // B/C column for this lane

    const float* __restrict__ aL = ldsA + (size_t)lo16 * K + khalf;
    const float* __restrict__ bp = B + (size_t)khalf * Ncols + n;

    v8f acc = {};
#pragma unroll 8
    for (int k0 = 0; k0 < K; k0 += 4) {
        v2f a, b;
        a.x = aL[k0];                            // ds_load_b64
        a.y = aL[k0 + 1];
        b.x = bp[(size_t)k0 * Ncols];            // global, L2-resident weights
        b.y = bp[(size_t)(k0 + 1) * Ncols];
        acc = __builtin_amdgcn_wmma_f32_16x16x4_f32(
            /*neg_a=*/false, a, /*neg_b=*/false, b,
            /*c_mod=*/(short)0, acc, /*reuse_a=*/false, /*reuse_b=*/false);
    }

    const int mrow = mtile * 16 + (lane >> 4) * 8;
    float* __restrict__ cp = C + (size_t)mrow * Ncols + n;
#pragma unroll
    for (int r = 0; r < 8; ++r) cp[(size_t)r * Ncols] = acc[r];
}

// ---------------------------------------------------------------------------
// alpha_s[n,h] = <feat[n,h,:], a_s[h,:]>, alpha_d likewise. One thread/(n,h).
// ---------------------------------------------------------------------------
__global__ __launch_bounds__(256)
void alpha_kernel(const float* __restrict__ feat,
                  const float* __restrict__ a_s, const float* __restrict__ a_d,
                  float* __restrict__ as_out, float* __restrict__ ad_out,
                  int N, int D) {
    int tid = blockIdx.x * blockDim.x + threadIdx.x;
    if (tid >= N * HEADS) return;
    int nidx = tid / HEADS, h = tid - nidx * HEADS;
    const float* f  = feat + (size_t)nidx * HEADS * D + (size_t)h * D;
    const float* vs = a_s + h * D;
    const float* vd = a_d + h * D;
    float ss = 0.f, sd = 0.f;
    for (int d = 0; d < D; ++d) { float x = f[d]; ss = fmaf(x, vs[d], ss); sd = fmaf(x, vd[d], sd); }
    as_out[tid] = ss;
    ad_out[tid] = sd;
}

__device__ __forceinline__ float leaky_relu(float x) {
    return x > 0.f ? x : x * LEAKY;
}

// Monotonic float atomic max (targets initialized to -inf).
__device__ __forceinline__ void atomicMaxF(float* addr, float v) {
    if (v >= 0.f) atomicMax((int*)addr, __float_as_int(v));
    else          atomicMin((unsigned int*)addr, (unsigned int)__float_as_int(v));
}

// ---------------------------------------------------------------------------
// Pass 1 over edges: segment max of leaky_relu(a_s[src] + a_d[dst]) by dst.
// ---------------------------------------------------------------------------
__global__ __launch_bounds__(256)
void edge_max_kernel(const float* __restrict__ as_, const float* __restrict__ ad_,
                     const int* __restrict__ src, const int* __restrict__ dst,
                     float* __restrict__ m) {
    int tid = blockIdx.x * blockDim.x + threadIdx.x;
    if (tid >= N_EDGES * HEADS) return;
    int e = tid / HEADS, h = tid - e * HEADS;
    int s = src[e], d = dst[e];
    float ev = leaky_relu(as_[(size_t)s * HEADS + h] + ad_[(size_t)d * HEADS + h]);
    atomicMaxF(m + (size_t)d * HEADS + h, ev);
}

// ---------------------------------------------------------------------------
// Pass 2: ex = exp(e - m[dst]); denom[dst] += ex (segment sum).
// ---------------------------------------------------------------------------
__global__ __launch_bounds__(256)
void edge_exp_kernel(const float* __restrict__ as_, const float* __restrict__ ad_,
                     const int* __restrict__ src, const int* __restrict__ dst,
                     const float* __restrict__ m,
                     float* __restrict__ ex, float* __restrict__ denom) {
    int tid = blockIdx.x * blockDim.x + threadIdx.x;
    if (tid >= N_EDGES * HEADS) return;
    int e = tid / HEADS, h = tid - e * HEADS;
    int s = src[e], d = dst[e];
    float ev = leaky_relu(as_[(size_t)s * HEADS + h] + ad_[(size_t)d * HEADS + h]);
    float x = __expf(ev - m[(size_t)d * HEADS + h]);
    ex[tid] = x;
    atomicAdd(denom + (size_t)d * HEADS + h, x);
}

// ---------------------------------------------------------------------------
// Pass 3: out[dst, h, :] += feat[src, h, :] * ex/(denom[dst,h]+eps).
// One thread per (edge, float4-of-features); D % 4 == 0, so all four scalars
// share one head -> one alpha, one b128 load, four f32 atomics.
// ---------------------------------------------------------------------------
__global__ __launch_bounds__(256)
void aggregate_kernel(const float* __restrict__ feat, const float* __restrict__ ex,
                      const float* __restrict__ denom,
                      const int* __restrict__ src, const int* __restrict__ dst,
                      float* __restrict__ out, int D) {
    const int F = HEADS * D;
    const int Q = F >> 2;                       // float4 groups per edge
    long long tid = (long long)blockIdx.x * blockDim.x + threadIdx.x;
    long long total = (long long)N_EDGES * Q;
    if (tid >= total) return;
    long long e = tid / Q;
    int q = (int)(tid - e * Q);
    int f = q << 2;
    int h = f / D;
    int s = src[e], d = dst[e];
    float alpha = ex[e * HEADS + h] / (denom[(size_t)d * HEADS + h] + 1e-16f);
    const float4 v = *(const float4*)(feat + (size_t)s * F + f);
    float* o = out + (size_t)d * F + f;
    atomicAdd(o + 0, v.x * alpha);
    atomicAdd(o + 1, v.y * alpha);
    atomicAdd(o + 2, v.z * alpha);
    atomicAdd(o + 3, v.w * alpha);
}

// y = elu(y + bias[f % F]), in place.
__global__ __launch_bounds__(256)
void elu_bias_kernel(float* __restrict__ y, const float* __restrict__ bias, int F, long long total) {
    long long tid = (long long)blockIdx.x * blockDim.x + threadIdx.x;
    if (tid >= total) return;
    float v = y[tid] + bias[tid % F];
    y[tid] = v > 0.f ? v : expm1f(v);
}

__global__ __launch_bounds__(256)
void fill_f32(float* __restrict__ p, float v, long long n) {
    long long i = (long long)blockIdx.x * blockDim.x + threadIdx.x;
    if (i < n) p[i] = v;
}

// ---------------------------------------------------------------------------
// Final: mean over heads of agg1[n,8,64], + b1, log_softmax over 64 classes.
// One wave32 per node; each lane owns 2 columns; shuffle reduction (wave32).
// ---------------------------------------------------------------------------
__global__ __launch_bounds__(256)
void finalize_kernel(const float* __restrict__ agg, const float* __restrict__ b1,
                     float* __restrict__ out) {
    int node = blockIdx.x * 8 + (threadIdx.x >> 5);
    int lane = threadIdx.x & 31;
    if (node >= N_NODES) return;
    const float* p = agg + (size_t)node * HEADS * N_CLASS;
    float v0 = 0.f, v1 = 0.f;
#pragma unroll
    for (int h = 0; h < HEADS; ++h) {
        v0 += p[h * N_CLASS + lane];
        v1 += p[h * N_CLASS + lane + 32];
    }
    v0 = v0 * (1.f / HEADS) + b1[lane];
    v1 = v1 * (1.f / HEADS) + b1[lane + 32];
    float mx = fmaxf(v0, v1);
#pragma unroll
    for (int o = 16; o > 0; o >>= 1) mx = fmaxf(mx, __shfl_xor(mx, o));
    float se = expf(v0 - mx) + expf(v1 - mx);
#pragma unroll
    for (int o = 16; o > 0; o >>= 1) se += __shfl_xor(se, o);
    float lse = mx + logf(se);
    out[(size_t)node * N_CLASS + lane]      = v0 - lse;
    out[(size_t)node * N_CLASS + lane + 32] = v1 - lse;
}

// ---------------------------------------------------------------------------

static inline long long cdiv_ll(long long a, long long b) { return (a + b - 1) / b; }

extern "C" void kernel_launch(void* const* d_in, const int* in_sizes, int n_in,
                              void* d_out, int out_size, void* d_ws, size_t ws_size,
                              hipStream_t stream) {
    const float* x      = (const float*)d_in[0];
    const int*   ei     = (const int*)d_in[1];
    const float* W0     = (const float*)d_in[2];
    const float* a_src0 = (const float*)d_in[3];
    const float* a_dst0 = (const float*)d_in[4];
    const float* b0     = (const float*)d_in[5];
    const float* W1     = (const float*)d_in[6];
    const float* a_src1 = (const float*)d_in[7];
    const float* a_dst1 = (const float*)d_in[8];
    const float* b1     = (const float*)d_in[9];
    float* out = (float*)d_out;

    const int* src = ei;
    const int* dst = ei + N_EDGES;

    // ---- workspace bump allocator (256B aligned) ----
    char* ws = (char*)d_ws;
    size_t off = 0;
    auto alloc = [&](size_t nfloats) -> float* {
        float* p = (float*)(ws + off);
        off += ((nfloats * sizeof(float)) + 255) & ~(size_t)255;
        return p;
    };

    float* agg0 = alloc((size_t)N_NODES * N_HID);   // persists: becomes layer-1 input
    size_t mark = off;                              // layer-scoped temporaries below

    // ================= Layer 0 (D=16, concat -> [N,128]) =================
    {
        const int D = N_HID / HEADS;                        // 16
        float* h0  = alloc((size_t)N_NODES * N_HID);        // x @ W0
        float* as0 = alloc((size_t)N_NODES * HEADS);
        float* ad0 = alloc((size_t)N_NODES * HEADS);
        float* m0  = alloc((size_t)N_NODES * HEADS);
        float* dn0 = alloc((size_t)N_NODES * HEADS);
        float* ex0 = alloc((size_t)N_EDGES * HEADS);

        // GEMM0: [50000,256] @ [256,128]; 3125 m-tiles, 8 n-tiles per block
        gemm_wmma_f32<<<dim3(N_NODES / 16, N_HID / 128), 256, 0, stream>>>(
            x, W0, h0, N_FEAT, N_HID);

        alpha_kernel<<<cdiv_ll((long long)N_NODES * HEADS, 256), 256, 0, stream>>>(
            h0, a_src0, a_dst0, as0, ad0, N_NODES, D);

        fill_f32<<<cdiv_ll((long long)N_NODES * HEADS, 256), 256, 0, stream>>>(
            m0, -INFINITY, (long long)N_NODES * HEADS);
        fill_f32<<<cdiv_ll((long long)N_NODES * HEADS, 256), 256, 0, stream>>>(
            dn0, 0.f, (long long)N_NODES * HEADS);
        fill_f32<<<cdiv_ll((long long)N_NODES * N_HID, 256), 256, 0, stream>>>(
            agg0, 0.f, (long long)N_NODES * N_HID);

        edge_max_kernel<<<cdiv_ll((long long)N_EDGES * HEADS, 256), 256, 0, stream>>>(
            as0, ad0, src, dst, m0);
        edge_exp_kernel<<<cdiv_ll((long long)N_EDGES * HEADS, 256), 256, 0, stream>>>(
            as0, ad0, src, dst, m0, ex0, dn0);
        aggregate_kernel<<<cdiv_ll((long long)N_EDGES * (N_HID / 4), 256), 256, 0, stream>>>(
            h0, ex0, dn0, src, dst, agg0, D);

        // h = elu(agg0 + b0)  (in place; agg0 is now the layer-1 input)
        elu_bias_kernel<<<cdiv_ll((long long)N_NODES * N_HID, 256), 256, 0, stream>>>(
            agg0, b0, N_HID, (long long)N_NODES * N_HID);
    }

    // ================= Layer 1 (D=64, mean over heads -> [N,64]) =========
    off = mark;  // layer-0 temporaries are dead; reuse the region
    {
        const int D = N_CLASS;                              // 64
        const int F = HEADS * N_CLASS;                      // 512
        float* h1  = alloc((size_t)N_NODES * F);            // agg0 @ W1
        float* as1 = alloc((size_t)N_NODES * HEADS);
        float* ad1 = alloc((size_t)N_NODES * HEADS);
        float* m1  = alloc((size_t)N_NODES * HEADS);
        float* dn1 = alloc((size_t)N_NODES * HEADS);
        float* ex1 = alloc((size_t)N_EDGES * HEADS);
        float* agg1 = alloc((size_t)N_NODES * F);

        // GEMM1: [50000,128] @ [128,512]; 3125 m-tiles x 4 block-rows of 8 n-tiles
        gemm_wmma_f32<<<dim3(N_NODES / 16, F / 128), 256, 0, stream>>>(
            agg0, W1, h1, N_HID, F);

        alpha_kernel<<<cdiv_ll((long long)N_NODES * HEADS, 256), 256, 0, stream>>>(
            h1, a_src1, a_dst1, as1, ad1, N_NODES, D);

        fill_f32<<<cdiv_ll((long long)N_NODES * HEADS, 256), 256, 0, stream>>>(
            m1, -INFINITY, (long long)N_NODES * HEADS);
        fill_f32<<<cdiv_ll((long long)N_NODES * HEADS, 256), 256, 0, stream>>>(
            dn1, 0.f, (long long)N_NODES * HEADS);
        fill_f32<<<cdiv_ll((long long)N_NODES * F, 256), 256, 0, stream>>>(
            agg1, 0.f, (long long)N_NODES * F);

        edge_max_kernel<<<cdiv_ll((long long)N_EDGES * HEADS, 256), 256, 0, stream>>>(
            as1, ad1, src, dst, m1);
        edge_exp_kernel<<<cdiv_ll((long long)N_EDGES * HEADS, 256), 256, 0, stream>>>(
            as1, ad1, src, dst, m1, ex1, dn1);
        aggregate_kernel<<<cdiv_ll((long long)N_EDGES * (F / 4), 256), 256, 0, stream>>>(
            h1, ex1, dn1, src, dst, agg1, D);

        // mean over heads + b1 + log_softmax -> out [50000, 64]
        finalize_kernel<<<cdiv_ll(N_NODES, 8), 256, 0, stream>>>(agg1, b1, out);
    }

    (void)in_sizes; (void)n_in; (void)out_size; (void)ws_size;
}